// HistoryEmbTable_44899588112506
// MI455X (gfx1250) — compile-verified
//
#include <hip/hip_runtime.h>
#include <hip/hip_bf16.h>

// HistoryEmbTable: emb.at[push_idx].set(x); out = emb_new[pull_idx]
// emb: [60M,1] f32 (immutable input), x: [4M,1] f32, idx: [4M] i32.
//
// Roofline: zero FLOPs -> pure memory op. Sparse-delta hash table in d_ws
// (96 MB, half L2-resident on the 192MB-L2 MI455X) instead of materializing
// the 240 MB updated table (~300-400 MB of DRAM traffic vs ~1 GB, i.e.
// ~15us vs ~43us at 23.3 TB/s).
//   keys[2^23] : u32, EMPTY=0xFFFFFFFF, claimed via global_atomic_cmpswap_b32
//   vals[2^23] : u64 packed ((rank+1)<<32 | f32bits), global_atomic_max_u64
//                -> deterministic last-write-wins for duplicate push indices
// Lookup probes the hash; on EMPTY (~93.5% of pulls) it falls back to a
// non-temporal gather from the original emb table. All one-shot streams
// (idx/x/out/emb) are NT so L2 stays dedicated to the hash table.

#define HASH_LOG2   23
#define HASH_SLOTS  (1u << HASH_LOG2)
#define HASH_MASK   (HASH_SLOTS - 1u)
#define EMPTY_KEY   0xFFFFFFFFu

__device__ __forceinline__ unsigned hash_fn(unsigned k) {
    return (k * 2654435761u) & HASH_MASK;   // Knuth multiplicative hash
}

// ---- hash path kernels -----------------------------------------------------

// Vectorized init: 16B stores (global_store_b128). keys/vals are reused by the
// insert+lookup kernels, so keep regular-temporal stores to seed L2.
__global__ void hemb_init_kernel(uint4* __restrict__ keys4,
                                 ulonglong2* __restrict__ vals2,
                                 unsigned nkeys4, unsigned nvals2) {
    unsigned i      = blockIdx.x * blockDim.x + threadIdx.x;
    unsigned stride = gridDim.x * blockDim.x;
    const uint4      kinit = make_uint4(EMPTY_KEY, EMPTY_KEY, EMPTY_KEY, EMPTY_KEY);
    const ulonglong2 vinit = {0ull, 0ull};
    for (unsigned j = i; j < nkeys4; j += stride) keys4[j] = kinit;
    for (unsigned j = i; j < nvals2; j += stride) vals2[j] = vinit;
}

__global__ void hemb_insert_kernel(const int* __restrict__ push_idx,
                                   const float* __restrict__ x,
                                   unsigned* __restrict__ keys,
                                   unsigned long long* __restrict__ vals,
                                   int n) {
    int i = blockIdx.x * blockDim.x + threadIdx.x;
    if (i >= n) return;
    unsigned k = (unsigned)__builtin_nontemporal_load(&push_idx[i]);
    float    v = __builtin_nontemporal_load(&x[i]);
    // rank+1 in high bits: atomicMax picks the highest push rank -> last write
    // wins deterministically (ranks are unique, so high bits always differ).
    unsigned long long packed =
        ((unsigned long long)(unsigned)(i + 1) << 32) |
        (unsigned long long)__float_as_uint(v);
    unsigned slot = hash_fn(k);
    for (;;) {
        unsigned prev = atomicCAS(&keys[slot], EMPTY_KEY, k);
        if (prev == EMPTY_KEY || prev == k) {
            atomicMax(&vals[slot], packed);
            return;
        }
        slot = (slot + 1u) & HASH_MASK;   // linear probe (keys never change once set)
    }
}

__global__ void hemb_lookup_kernel(const int* __restrict__ pull_idx,
                                   const float* __restrict__ emb,
                                   const unsigned* __restrict__ keys,
                                   const unsigned long long* __restrict__ vals,
                                   float* __restrict__ out,
                                   int n) {
    int i = blockIdx.x * blockDim.x + threadIdx.x;
    if (i >= n) return;
    unsigned k    = (unsigned)__builtin_nontemporal_load(&pull_idx[i]);
    unsigned slot = hash_fn(k);
    float r;
    for (;;) {
        unsigned kk = keys[slot];
        if (kk == k) {                       // pushed entry: take hash value
            unsigned long long p = vals[slot];
            r = __uint_as_float((unsigned)(p & 0xFFFFFFFFull));
            break;
        }
        if (kk == EMPTY_KEY) {               // not pushed: NT gather from emb
            r = __builtin_nontemporal_load(&emb[k]);
            break;
        }
        slot = (slot + 1u) & HASH_MASK;
    }
    __builtin_nontemporal_store(r, &out[i]);
}

// ---- fallback: materialize full table in d_ws ------------------------------

__global__ void hemb_scatter_kernel(const int* __restrict__ push_idx,
                                    const float* __restrict__ x,
                                    float* __restrict__ table, int n) {
    int i = blockIdx.x * blockDim.x + threadIdx.x;
    if (i >= n) return;
    __builtin_nontemporal_store(x[i], &table[push_idx[i]]);
}

__global__ void hemb_gather_kernel(const int* __restrict__ pull_idx,
                                   const float* __restrict__ table,
                                   float* __restrict__ out, int n) {
    int i = blockIdx.x * blockDim.x + threadIdx.x;
    if (i >= n) return;
    float r = __builtin_nontemporal_load(&table[pull_idx[i]]);
    __builtin_nontemporal_store(r, &out[i]);
}

// ---- last-resort: gather from original emb (pushes dropped) ----------------

__global__ void hemb_gather_plain_kernel(const int* __restrict__ pull_idx,
                                         const float* __restrict__ emb,
                                         float* __restrict__ out, int n) {
    int i = blockIdx.x * blockDim.x + threadIdx.x;
    if (i >= n) return;
    float r = __builtin_nontemporal_load(&emb[pull_idx[i]]);
    __builtin_nontemporal_store(r, &out[i]);
}

extern "C" void kernel_launch(void* const* d_in, const int* in_sizes, int n_in,
                              void* d_out, int out_size, void* d_ws, size_t ws_size,
                              hipStream_t stream) {
    const float* emb      = (const float*)d_in[0];
    const float* x        = (const float*)d_in[1];
    const int*   push_idx = (const int*)d_in[2];
    const int*   pull_idx = (const int*)d_in[3];
    float*       out      = (float*)d_out;

    const size_t num_emb = (size_t)in_sizes[0];   // EMB_DIM == 1
    const int    n       = in_sizes[2];           // N_IDX

    const int    BLK    = 256;                    // 8 wave32 waves per block
    const int    nblk   = (n + BLK - 1) / BLK;

    const size_t hash_bytes = (size_t)HASH_SLOTS * (sizeof(unsigned long long) + sizeof(unsigned));

    if (ws_size >= hash_bytes && (size_t)n <= (size_t)(HASH_SLOTS / 2)) {
        // --- sparse-delta hash path (expected path) ---
        unsigned long long* vals = (unsigned long long*)d_ws;            // 64 MB, 8B-aligned
        unsigned*           keys = (unsigned*)(vals + HASH_SLOTS);       // 32 MB

        hemb_init_kernel<<<4096, BLK, 0, stream>>>((uint4*)keys, (ulonglong2*)vals,
                                                   HASH_SLOTS / 4, HASH_SLOTS / 2);
        hemb_insert_kernel<<<nblk, BLK, 0, stream>>>(push_idx, x, keys, vals, n);
        hemb_lookup_kernel<<<nblk, BLK, 0, stream>>>(pull_idx, emb, keys, vals, out, n);
    } else if (ws_size >= num_emb * sizeof(float)) {
        // --- materialize updated table in workspace ---
        float* table = (float*)d_ws;
        hipMemcpyAsync(table, emb, num_emb * sizeof(float),
                       hipMemcpyDeviceToDevice, stream);
        hemb_scatter_kernel<<<nblk, BLK, 0, stream>>>(push_idx, x, table, n);
        hemb_gather_kernel<<<nblk, BLK, 0, stream>>>(pull_idx, table, out, n);
    } else {
        // --- workspace too small for any correct plan: best-effort gather ---
        hemb_gather_plain_kernel<<<nblk, BLK, 0, stream>>>(pull_idx, emb, out, n);
    }
}